// Encoder_20779051778614
// MI455X (gfx1250) — compile-verified
//
#include <hip/hip_runtime.h>
#include <math.h>

// ---------------- problem constants ----------------
constexpr int Bn  = 16;
constexpr int Dd  = 2;
constexpr int NNn = 100;
constexpr int N2n = 50;
constexpr int NAn = Dd + NNn;   // 102
constexpr int Hn  = 128;
constexpr int HEn = 64;
constexpr int Ln  = 3;
constexpr float SLOPEc = 0.2f;
constexpr float EPSc   = 1e-5f;

typedef float v2f __attribute__((ext_vector_type(2)));
typedef float v8f __attribute__((ext_vector_type(8)));

// ---------------- small utility kernels ----------------
__global__ void zero_ws_kernel(float* p, int n) {
    int i = blockIdx.x * blockDim.x + threadIdx.x;
    if (i < n) p[i] = 0.0f;
}

// 5 scalar moments of edge_attr (rows of 2): S0,S1,Q00,Q11,Q01
__global__ void edge_moments_kernel(const float* __restrict__ e, float* __restrict__ st) {
    const size_t n = (size_t)Bn * NAn * NAn;
    float s0 = 0, s1 = 0, q00 = 0, q11 = 0, q01 = 0;
    for (size_t i = (size_t)blockIdx.x * blockDim.x + threadIdx.x; i < n;
         i += (size_t)gridDim.x * blockDim.x) {
        float a = e[2 * i], b = e[2 * i + 1];
        s0 += a; s1 += b; q00 += a * a; q11 += b * b; q01 += a * b;
    }
    for (int d = 16; d > 0; d >>= 1) {
        s0  += __shfl_down(s0, d);  s1  += __shfl_down(s1, d);
        q00 += __shfl_down(q00, d); q11 += __shfl_down(q11, d);
        q01 += __shfl_down(q01, d);
    }
    if ((threadIdx.x & 31) == 0) {
        atomicAdd(&st[0], s0);  atomicAdd(&st[1], s1);
        atomicAdd(&st[2], q00); atomicAdd(&st[3], q11);
        atomicAdd(&st[4], q01);
    }
}

// Collapse BN(edge@W) @ we  into  e0*M0[h] + e1*M1[h] + bias[h].
// blockIdx.x = g*9 + l*3 + t  (g: 0=d,1=r; t: 0=all,1=pick,2=deli), thread = h.
__global__ void precompute_M_kernel(const float* __restrict__ stats,
                                    const float* __restrict__ W3, const float* __restrict__ W4,
                                    const float* __restrict__ b3g, const float* __restrict__ b3b,
                                    const float* __restrict__ b4g, const float* __restrict__ b4b,
                                    const float* __restrict__ Wga, const float* __restrict__ Wgp,
                                    const float* __restrict__ Wgd,
                                    float* __restrict__ Mout) {
    const int h = threadIdx.x;
    const int idx = blockIdx.x;
    const int g = idx / 9, l = (idx / 3) % 3, t = idx % 3;
    const float* WE = (g == 0) ? W3 : W4;
    const float* gg = (g == 0) ? b3g : b4g;
    const float* bb = (g == 0) ? b3b : b4b;
    const float* Wg = (t == 0) ? Wga : ((t == 1) ? Wgp : Wgd);
    const float* we = Wg + (size_t)l * (2 * Hn + HEn) * Hn + (size_t)2 * Hn * Hn; // rows 2H..2H+HE
    const float* st = stats + g * 8;
    const float invN = 1.0f / (float)((size_t)Bn * NAn * NAn);
    float M0 = 0.f, M1 = 0.f, bias = 0.f;
    for (int c = 0; c < HEn; c++) {
        float w0 = WE[c], w1 = WE[HEn + c];
        float mean = (st[0] * w0 + st[1] * w1) * invN;
        float ey2  = (st[2] * w0 * w0 + 2.f * st[4] * w0 * w1 + st[3] * w1 * w1) * invN;
        float var  = ey2 - mean * mean;
        float scale = gg[c] * rsqrtf(var + EPSc);
        float shift = bb[c] - mean * scale;
        float wech  = we[(size_t)c * Hn + h];
        M0 += w0 * scale * wech;
        M1 += w1 * scale * wech;
        bias += shift * wech;
    }
    float* o = Mout + (size_t)idx * 384;
    o[h] = M0; o[Hn + h] = M1; o[2 * Hn + h] = bias;
}

// ---------------- node embeddings + BN ----------------
__device__ __forceinline__ float node_feat(const float* x, const float* dem, const float* tw,
                                           int b, int n, int f) {
    switch (f) {
        case 0:  return x[((size_t)b * NAn + n) * 2 + 0];
        case 1:  return x[((size_t)b * NAn + n) * 2 + 1];
        case 2:  return dem[(size_t)b * NAn + n];
        case 3:  return tw[((size_t)b * NAn + n) * 2 + 0];
        default: return tw[((size_t)b * NAn + n) * 2 + 1];
    }
}

// single block, 128 threads (one per output channel h); two passes (stats, normalize)
__global__ void embed_bn_kernel(const float* __restrict__ x, const float* __restrict__ dem,
                                const float* __restrict__ tw, const float* __restrict__ W,
                                int nodeOff, int pairOff, int rpb,
                                const float* __restrict__ gma, const float* __restrict__ bta,
                                float* __restrict__ outd, float* __restrict__ outr, int outOff) {
    const int h = threadIdx.x;
    const int rows = Bn * rpb;
    float sum = 0.f, sq = 0.f;
    for (int r = 0; r < rows; r++) {
        int b = r / rpb, p = r % rpb;
        float y = 0.f;
        for (int f = 0; f < 5; f++) y += node_feat(x, dem, tw, b, nodeOff + p, f) * W[f * Hn + h];
        if (pairOff >= 0)
            for (int f = 0; f < 5; f++) y += node_feat(x, dem, tw, b, pairOff + p, f) * W[(5 + f) * Hn + h];
        sum += y; sq += y * y;
    }
    const float mean = sum / rows;
    const float var  = sq / rows - mean * mean;
    const float sc = gma[h] * rsqrtf(var + EPSc);
    const float sh = bta[h] - mean * sc;
    for (int r = 0; r < rows; r++) {
        int b = r / rpb, p = r % rpb;
        float y = 0.f;
        for (int f = 0; f < 5; f++) y += node_feat(x, dem, tw, b, nodeOff + p, f) * W[f * Hn + h];
        if (pairOff >= 0)
            for (int f = 0; f < 5; f++) y += node_feat(x, dem, tw, b, pairOff + p, f) * W[(5 + f) * Hn + h];
        float o = y * sc + sh;
        size_t oi = ((size_t)b * NAn + outOff + p) * Hn + h;
        outd[oi] = o;
        outr[oi] = o;
    }
}

// ---------------- fp32 WMMA GEMM  C = gather(A) @ W  (+epilogue) ----------------
// one wave per 16x16 tile; K=128 in steps of 4 via v_wmma_f32_16x16x4_f32.
// A logical row m: b=m/rpb, loc=m%rpb, data at A + b*bStride + (rowOff+loc)*128.
// mode: 0 none, 1 +bias,ReLU, 2 +bias,+residual.
__global__ void wmma_gemm_f32_kernel(const float* __restrict__ A, int rpb, int rowOff, int bStride,
                                     const float* __restrict__ W, float* __restrict__ C,
                                     const float* __restrict__ bias,
                                     const float* __restrict__ resid, int mode) {
    const int lane = threadIdx.x;        // 0..31, wave32
    const int m0 = blockIdx.x * 16;
    const int n0 = blockIdx.y * 16;
    const int half = lane >> 4;          // 0 / 1
    const int l15  = lane & 15;
    // A fragment row for this lane (ISA 7.12.2: 32-bit A 16x4, lane halves = K pairs)
    const int mrow = m0 + l15;
    const int bb = mrow / rpb;
    const int loc = mrow - bb * rpb;
    const float* arow = A + (size_t)bb * bStride + (size_t)(rowOff + loc) * Hn;
    const int ncol = n0 + l15;
    v8f acc = {};
    for (int k0 = 0; k0 < Hn; k0 += 4) {
        const int ka = k0 + half * 2;
        v2f af, bf;
        af.x = arow[ka];
        af.y = arow[ka + 1];
        bf.x = W[(size_t)ka * Hn + ncol];
        bf.y = W[(size_t)(ka + 1) * Hn + ncol];
        acc = __builtin_amdgcn_wmma_f32_16x16x4_f32(false, af, false, bf,
                                                    (short)0, acc, false, false);
    }
    // C layout: VGPR v -> row m0+v (lanes 0-15) / m0+v+8 (lanes 16-31), col n0+(lane&15)
#pragma unroll
    for (int v = 0; v < 8; v++) {
        const int row = m0 + v + half * 8;
        const int col = n0 + l15;
        float val = acc[v];
        if (mode == 1) {
            val += bias[col];
            val = val > 0.f ? val : 0.f;
        } else if (mode == 2) {
            val += bias[col] + resid[(size_t)row * Hn + col];
        }
        C[(size_t)row * Hn + col] = val;
    }
}

// ---------------- attention prop (online softmax over i) ----------------
// block = (j, b), thread = h.  alpha[b,i,j,h] = SJ[b,j,h]+SI[b,i,h]+e0*M0+e1*M1+bias
// out[b, outOff+j, h] (=|+=) sum_i softmax_i(alpha) * XK[b,i,h]
__global__ void attn_prop_kernel(const float* __restrict__ SI, const float* __restrict__ SJ,
                                 const float* __restrict__ XK, int rpn,
                                 const float* __restrict__ eattr, const int* __restrict__ mask,
                                 int off, const float* __restrict__ Mblk,
                                 float* __restrict__ out, int outOff, int n, int accum) {
    const int h = threadIdx.x;
    const int j = blockIdx.x;
    const int b = blockIdx.y;
    const float m0h = Mblk[h], m1h = Mblk[Hn + h], bh = Mblk[2 * Hn + h];
    const float sjv = SJ[((size_t)b * rpn + j) * Hn + h];
    const size_t ebase = (size_t)b * NAn * NAn;
    float rmax = -3.0e38f, rsum = 0.f, racc = 0.f;
    for (int i = 0; i < n; i++) {
        const size_t eidx = ebase + (size_t)(off + i) * NAn + (off + j);
        if (mask[eidx] == 0) continue;                // block-uniform branch
        const float e0 = eattr[2 * eidx], e1 = eattr[2 * eidx + 1];
        const float siv = SI[((size_t)b * rpn + i) * Hn + h];
        const float xkv = XK[((size_t)b * rpn + i) * Hn + h];
        float v = sjv + siv + e0 * m0h + e1 * m1h + bh;
        v = v >= 0.f ? v : SLOPEc * v;                // leaky relu
        if (v > rmax) {
            const float sc = __expf(rmax - v);
            rsum = rsum * sc + 1.f;
            racc = racc * sc + xkv;
            rmax = v;
        } else {
            const float w = __expf(v - rmax);
            rsum += w;
            racc += w * xkv;
        }
    }
    const float o = racc / rsum;
    float* op = out + ((size_t)b * NAn + outOff + j) * Hn + h;
    if (accum) *op += o; else *op = o;
}

// ---------------- final FF batchnorm ----------------
__global__ void bn_stats_kernel(const float* __restrict__ y, int rows, float* __restrict__ ms) {
    const int h = threadIdx.x;
    float s = 0.f, q = 0.f;
    for (int r = 0; r < rows; r++) {
        float v = y[(size_t)r * Hn + h];
        s += v; q += v * v;
    }
    const float mean = s / rows;
    const float var  = q / rows - mean * mean;
    ms[h] = mean;
    ms[Hn + h] = rsqrtf(var + EPSc);
}

__global__ void bn_apply_kernel(const float* __restrict__ y, const float* __restrict__ ms,
                                const float* __restrict__ g, const float* __restrict__ b,
                                float* __restrict__ out) {
    const int h = threadIdx.x;
    const size_t r = blockIdx.x;
    const float v = y[r * Hn + h];
    out[r * Hn + h] = (v - ms[h]) * ms[Hn + h] * g[h] + b[h];
}

// ---------------- host launcher ----------------
extern "C" void kernel_launch(void* const* d_in, const int* in_sizes, int n_in,
                              void* d_out, int out_size, void* d_ws, size_t ws_size,
                              hipStream_t stream) {
    (void)in_sizes; (void)n_in; (void)out_size; (void)ws_size;
    const float* x   = (const float*)d_in[0];
    const float* dem = (const float*)d_in[1];
    const float* tw  = (const float*)d_in[2];
    const float* e_d = (const float*)d_in[3];
    const float* e_r = (const float*)d_in[4];
    const float* W0  = (const float*)d_in[5];
    const float* W1  = (const float*)d_in[6];
    const float* W2  = (const float*)d_in[7];
    const float* W3  = (const float*)d_in[8];
    const float* W4  = (const float*)d_in[9];
    const float* b0g = (const float*)d_in[10], *b0b = (const float*)d_in[11];
    const float* b1g = (const float*)d_in[12], *b1b = (const float*)d_in[13];
    const float* b2g = (const float*)d_in[14], *b2b = (const float*)d_in[15];
    const float* b3g = (const float*)d_in[16], *b3b = (const float*)d_in[17];
    const float* b4g = (const float*)d_in[18], *b4b = (const float*)d_in[19];
    const float* ffw1 = (const float*)d_in[20], *ffb1 = (const float*)d_in[21];
    const float* ffw2 = (const float*)d_in[22], *ffb2 = (const float*)d_in[23];
    const float* bng  = (const float*)d_in[24], *bnb  = (const float*)d_in[25];
    const float* Wvla = (const float*)d_in[26];
    const float* Wvlp = (const float*)d_in[27];
    const float* Wvld = (const float*)d_in[28];
    const float* Wga  = (const float*)d_in[29];
    const float* Wgp  = (const float*)d_in[30];
    const float* Wgd  = (const float*)d_in[31];
    /* d_in[32] = edge_index (unused) */
    const int* m_d = (const int*)d_in[33];
    const int* m_r = (const int*)d_in[34];
    float* out = (float*)d_out;
    float* ws  = (float*)d_ws;

    // workspace layout (float units)
    size_t off = 0;
    float* stats = ws + off; off += 16;                   // [g][S0,S1,Q00,Q11,Q01]
    float* Mblk  = ws + off; off += (size_t)18 * 384;     // [g][l][t][M0|M1|bias]
    const size_t SZX = (size_t)Bn * NAn * Hn;             // 208896
    const size_t SZP = (size_t)Bn * N2n * Hn;             // 102400
    float* xd0 = ws + off; off += SZX;
    float* xd1 = ws + off; off += SZX;
    float* xr0 = ws + off; off += SZX;
    float* xr1 = ws + off; off += SZX;
    float* xall = ws + off; off += SZX;
    float* SIa = ws + off; off += SZX;
    float* SJa = ws + off; off += SZX;
    float* pick = ws + off; off += SZP;
    float* SIp = ws + off; off += SZP;
    float* SJp = ws + off; off += SZP;
    float* deli = ws + off; off += SZP;
    float* SId = ws + off; off += SZP;
    float* SJd = ws + off; off += SZP;
    float* h1  = ws + off; off += SZX;
    float* yff = ws + off; off += SZX;
    float* bnms = ws + off; off += 256;

    auto gemm = [&](const float* A, int rpb, int rowOff, int bStride, const float* W,
                    float* C, int M, const float* bias, const float* resid, int mode) {
        dim3 grid(M / 16, Hn / 16);
        wmma_gemm_f32_kernel<<<grid, dim3(32), 0, stream>>>(A, rpb, rowOff, bStride, W, C,
                                                            bias, resid, mode);
    };

    // ---- edge BN stats (analytic, from 5 moments) + collapsed projections ----
    zero_ws_kernel<<<1, 32, 0, stream>>>(stats, 16);
    edge_moments_kernel<<<dim3(104), dim3(256), 0, stream>>>(e_d, stats);
    edge_moments_kernel<<<dim3(104), dim3(256), 0, stream>>>(e_r, stats + 8);
    precompute_M_kernel<<<dim3(18), dim3(128), 0, stream>>>(stats, W3, W4, b3g, b3b, b4g, b4b,
                                                            Wga, Wgp, Wgd, Mblk);

    // ---- node embeddings (dep / pickup / delivery), write both graph copies ----
    embed_bn_kernel<<<1, 128, 0, stream>>>(x, dem, tw, W0, 0,        -1,        Dd,  b0g, b0b, xd0, xr0, 0);
    embed_bn_kernel<<<1, 128, 0, stream>>>(x, dem, tw, W1, Dd,       Dd + N2n, N2n, b1g, b1b, xd0, xr0, Dd);
    embed_bn_kernel<<<1, 128, 0, stream>>>(x, dem, tw, W2, Dd + N2n, -1,       N2n, b2g, b2b, xd0, xr0, Dd + N2n);

    // ---- L graph-attention layers, two independent graph chains ----
    float* cur[2] = { xd0, xr0 };
    float* nxt[2] = { xd1, xr1 };
    const float* earr[2] = { e_d, e_r };
    const int* marr[2] = { m_d, m_r };

    for (int l = 0; l < Ln; l++) {
        const float* wva = Wvla + (size_t)l * Hn * Hn;
        const float* wvp = Wvlp + (size_t)l * Hn * Hn;
        const float* wvd = Wvld + (size_t)l * Hn * Hn;
        const float* ga = Wga + (size_t)l * (2 * Hn + HEn) * Hn;  // rows: wi | wj | we
        const float* gp = Wgp + (size_t)l * (2 * Hn + HEn) * Hn;
        const float* gd = Wgd + (size_t)l * (2 * Hn + HEn) * Hn;
        for (int g = 0; g < 2; g++) {
            const float* xc = cur[g];
            float* xn = nxt[g];
            // value projections
            gemm(xc, NAn, 0,        NAn * Hn, wva, xall, Bn * NAn, nullptr, nullptr, 0);
            gemm(xc, N2n, Dd,       NAn * Hn, wvp, pick, Bn * N2n, nullptr, nullptr, 0);
            gemm(xc, N2n, Dd + N2n, NAn * Hn, wvd, deli, Bn * N2n, nullptr, nullptr, 0);
            // attention logits: SJ = xk@wi (indexed by j), SI = xk@wj (indexed by i)
            gemm(xall, NAn, 0, NAn * Hn, ga,            SJa, Bn * NAn, nullptr, nullptr, 0);
            gemm(xall, NAn, 0, NAn * Hn, ga + Hn * Hn,  SIa, Bn * NAn, nullptr, nullptr, 0);
            gemm(pick, N2n, 0, N2n * Hn, gp,            SJp, Bn * N2n, nullptr, nullptr, 0);
            gemm(pick, N2n, 0, N2n * Hn, gp + Hn * Hn,  SIp, Bn * N2n, nullptr, nullptr, 0);
            gemm(deli, N2n, 0, N2n * Hn, gd,            SJd, Bn * N2n, nullptr, nullptr, 0);
            gemm(deli, N2n, 0, N2n * Hn, gd + Hn * Hn,  SId, Bn * N2n, nullptr, nullptr, 0);
            // softmax-aggregate; pick/deli accumulate into xn via stream ordering
            attn_prop_kernel<<<dim3(NAn, Bn), 128, 0, stream>>>(
                SIa, SJa, xall, NAn, earr[g], marr[g], 0,
                Mblk + (size_t)(g * 9 + l * 3 + 0) * 384, xn, 0, NAn, 0);
            attn_prop_kernel<<<dim3(N2n, Bn), 128, 0, stream>>>(
                SIp, SJp, pick, N2n, earr[g], marr[g], Dd,
                Mblk + (size_t)(g * 9 + l * 3 + 1) * 384, xn, Dd, N2n, 1);
            attn_prop_kernel<<<dim3(N2n, Bn), 128, 0, stream>>>(
                SId, SJd, deli, N2n, earr[g], marr[g], Dd + N2n,
                Mblk + (size_t)(g * 9 + l * 3 + 2) * 384, xn, Dd + N2n, N2n, 1);
            // ping-pong
            float* t = cur[g]; cur[g] = nxt[g]; nxt[g] = t;
        }
    }

    // ---- feed-forward + residual + BN, straight into d_out ----
    for (int g = 0; g < 2; g++) {
        const float* xf = cur[g];
        gemm(xf, NAn, 0, NAn * Hn, ffw1, h1,  Bn * NAn, ffb1, nullptr, 1);  // ReLU
        gemm(h1, NAn, 0, NAn * Hn, ffw2, yff, Bn * NAn, ffb2, xf,      2);  // +residual
        bn_stats_kernel<<<1, 128, 0, stream>>>(yff, Bn * NAn, bnms);
        bn_apply_kernel<<<Bn * NAn, 128, 0, stream>>>(yff, bnms, bng, bnb, out + (size_t)g * SZX);
    }
}